// SCAESuite_17497696764055
// MI455X (gfx1250) — compile-verified
//
#include <hip/hip_runtime.h>
#include <hip/hip_bf16.h>

// Problem constants (from the reference)
#define B_SZ     2048
#define K_UP     32
#define K_DOWN   32
#define D_SZ     768
#define F_UP     24576
#define F_DOWN   24576
#define C_SZ     64

// Tiling
#define CHUNK    64             // D-chunk staged in LDS (double buffered)
#define RSTRIDE  (CHUNK + 4)    // 68 floats = 272B row stride: 16B aligned, bank-skewed (68 mod 64 = 4)
#define NCHUNK   (D_SZ / CHUNK) // 12

typedef float v2f __attribute__((ext_vector_type(2)));
typedef float v8f __attribute__((ext_vector_type(8)));

// ---------------------------------------------------------------------------
// Kernel 1: bias[f] = dot(w_enc_down[f,:], b_dec_up)   (one wave per row)
// ---------------------------------------------------------------------------
__global__ void scae_bias_kernel(const float* __restrict__ w_enc_down,
                                 const float* __restrict__ b_dec_up,
                                 float* __restrict__ bias) {
    int f    = blockIdx.x * 8 + (threadIdx.x >> 5);
    int lane = threadIdx.x & 31;
    float s = 0.f;
    const float* row = w_enc_down + (size_t)f * D_SZ;
    #pragma unroll 4
    for (int d = lane; d < D_SZ; d += 32)
        s += row[d] * b_dec_up[d];
    #pragma unroll
    for (int off = 16; off > 0; off >>= 1)
        s += __shfl_down(s, off, 32);
    if (lane == 0) bias[f] = s;
}

// ---------------------------------------------------------------------------
// Kernel 2: w_dec_upT[f][d] = w_dec_up[d][f]  (tiled 32x32 transpose)
// ---------------------------------------------------------------------------
__global__ void scae_transpose_kernel(const float* __restrict__ in,   // [D][F_UP]
                                      float* __restrict__ outT) {     // [F_UP][D]
    __shared__ float tile[32][33];
    int fb = blockIdx.x * 32;           // F_UP block
    int db = blockIdx.y * 32;           // D block
    int tx = threadIdx.x & 31;
    int ty = threadIdx.x >> 5;          // 8 waves -> 4 rows each
    #pragma unroll
    for (int r = ty; r < 32; r += 8)
        tile[r][tx] = in[(size_t)(db + r) * F_UP + fb + tx];
    __syncthreads();
    #pragma unroll
    for (int r = ty; r < 32; r += 8)
        outT[(size_t)(fb + r) * D_SZ + db + tx] = tile[tx][r];
}

// ---------------------------------------------------------------------------
// Kernel 3: main per-batch kernel. 256 threads (8 wave32) per batch.
//   waves 0..7 : tile = wave&3 (2x2 grid of 16x16 output tiles of V[32][32])
//                khalf = wave>>2 (splits each 64-wide D chunk into 2x32)
// Staging of the gathered rows for chunk ch+1 runs asynchronously
// (global_load_async_to_lds_b128, ASYNCcnt) while chunk ch feeds the WMMAs.
// ---------------------------------------------------------------------------
__global__ void __launch_bounds__(256)
scae_main_kernel(const float* __restrict__ up_vals,      // [B,32]
                 const float* __restrict__ w_dec_up,     // [D,F_UP]   (fallback path)
                 const float* __restrict__ w_enc_down,   // [F_DOWN,D]
                 const float* __restrict__ b_dec_up,     // [D]        (fallback path)
                 const int*   __restrict__ up_indices,   // [B,32]
                 const int*   __restrict__ down_indices, // [B,32]
                 const int*   __restrict__ connections,  // [F_DOWN,64]
                 float*       __restrict__ out,          // [B,32]
                 const float* __restrict__ wT,           // [F_UP,D] or null
                 const float* __restrict__ biasWs) {     // [F_DOWN] or null
    __shared__ float lds_down[2][32 * RSTRIDE];  // gathered down rows (double buffer)
    __shared__ float lds_up  [2][32 * RSTRIDE];  // gathered up rows   (double buffer)
    __shared__ float lds_V   [32 * 33];          // V[k][u], padded
    __shared__ int   lds_conn[32 * C_SZ];        // connection lists of down feats
    __shared__ int   lds_upidx[32];
    __shared__ int   lds_dnidx[32];
    __shared__ float lds_upvals[32];
    __shared__ float lds_contrib[32];

    const int b = blockIdx.x;
    const int t = threadIdx.x;

    if (t < 32) {
        lds_upidx[t]   = up_indices[(size_t)b * 32 + t];
        lds_dnidx[t]   = down_indices[(size_t)b * 32 + t];
        lds_upvals[t]  = up_vals[(size_t)b * 32 + t];
        lds_contrib[t] = 0.f;
    }
    for (int i = t; i < 32 * 33; i += 256) lds_V[i] = 0.f;
    __syncthreads();

    // Gather the 32 connection rows (each 64 contiguous ints)
    for (int i = t; i < 32 * C_SZ; i += 256) {
        int k = i >> 6, c = i & 63;
        lds_conn[i] = connections[(size_t)lds_dnidx[k] * C_SZ + c];
    }

    // Stage one D-chunk into LDS buffer `buf`.
    //   wT path: 4 async b128 issues per thread (2 per matrix) -> ASYNCcnt
    //   fallback: down matrix async (2 issues), up matrix synchronous strided gather
    auto stage = [&](int buf, int ch) {
        const int base = ch * CHUNK;
        for (int i = t; i < 32 * (CHUNK / 4); i += 256) {   // 512 items -> 2/thread
            const int r  = i >> 4;          // row 0..31   (CHUNK/4 == 16)
            const int c4 = (i & 15) << 2;   // col 0..60 step 4
            const float* dsrc = w_enc_down + (size_t)lds_dnidx[r] * D_SZ + base + c4;
            unsigned dofs = (unsigned)(uintptr_t)&lds_down[buf][r * RSTRIDE + c4];
            asm volatile("global_load_async_to_lds_b128 %0, %1, off"
                         :: "v"(dofs),
                            "v"((unsigned long long)(uintptr_t)dsrc)
                         : "memory");
            if (wT) {
                const float* usrc = wT + (size_t)lds_upidx[r] * D_SZ + base + c4;
                unsigned uofs = (unsigned)(uintptr_t)&lds_up[buf][r * RSTRIDE + c4];
                asm volatile("global_load_async_to_lds_b128 %0, %1, off"
                             :: "v"(uofs),
                                "v"((unsigned long long)(uintptr_t)usrc)
                             : "memory");
            } else {
                const int f = lds_upidx[r];  // strided column gather fallback
                float4 uv;
                uv.x = w_dec_up[(size_t)(base + c4 + 0) * F_UP + f];
                uv.y = w_dec_up[(size_t)(base + c4 + 1) * F_UP + f];
                uv.z = w_dec_up[(size_t)(base + c4 + 2) * F_UP + f];
                uv.w = w_dec_up[(size_t)(base + c4 + 3) * F_UP + f];
                *(float4*)&lds_up[buf][r * RSTRIDE + c4] = uv;
            }
        }
    };

    const int wave  = t >> 5;
    const int lane  = t & 31;
    const int tile  = wave & 3;
    const int khalf = wave >> 2;
    const int tm    = (tile >> 1) * 16;
    const int tn    = (tile & 1) * 16;
    const int mrow  = lane & 15;
    const int klane = (lane >> 4) * 2;     // lanes 16..31 hold K+2,K+3

    v8f acc = {0.f, 0.f, 0.f, 0.f, 0.f, 0.f, 0.f, 0.f};

    stage(0, 0);   // prologue: chunk 0 in flight

    for (int ch = 0; ch < NCHUNK; ++ch) {
        const int cur = ch & 1;
        if (ch + 1 < NCHUNK) {
            stage(cur ^ 1, ch + 1);  // overlap staging of next chunk with compute
            // In-order async completion: waiting to <= issues-per-chunk drains chunk ch
            if (wT) asm volatile("s_wait_asynccnt 0x4" ::: "memory");
            else    asm volatile("s_wait_asynccnt 0x2" ::: "memory");
        } else {
            asm volatile("s_wait_asynccnt 0x0" ::: "memory");
        }
        __syncthreads();   // chunk ch fully staged by all waves

        // 8 x v_wmma_f32_16x16x4_f32 per wave per chunk (K split over 2 waves)
        const int k0 = khalf * 32;
        #pragma unroll
        for (int k = 0; k < 32; k += 4) {
            v2f a  = *(const v2f*)&lds_down[cur][(tm + mrow) * RSTRIDE + k0 + k + klane];
            v2f bb = *(const v2f*)&lds_up  [cur][(tn + mrow) * RSTRIDE + k0 + k + klane];
            acc = __builtin_amdgcn_wmma_f32_16x16x4_f32(
                false, a, false, bb, (short)0, acc, false, false);
        }
        __syncthreads();   // protect buffer cur from being overwritten (chunk ch+2)
    }

    // Merge the two K-half partials of each tile into lds_V (ds_add_f32)
    #pragma unroll
    for (int r = 0; r < 8; ++r) {
        const int M = r + ((lane >> 4) << 3);  // VGPR r: lanes 0-15 M=r, 16-31 M=r+8
        const int N = lane & 15;
        atomicAdd(&lds_V[(tm + M) * 33 + tn + N], acc[r]);
    }
    __syncthreads();

    // Phase 2: contributions[k] = sum_u count(k,u) * V[k,u] * up_vals[u]
    {
        const int k  = t >> 3;    // 8 threads per k
        const int u0 = t & 7;
        float sum = 0.f;
        #pragma unroll
        for (int uu = 0; uu < 4; ++uu) {
            const int u      = u0 + (uu << 3);
            const int target = lds_upidx[u];
            int cnt = 0;
            #pragma unroll 8
            for (int c = 0; c < C_SZ; ++c) {
                const int cc = (c + (k << 4)) & 63;   // bank-rotated scan
                const int cv = lds_conn[(k << 6) + cc];
                cnt += (cv == target && cv != -1) ? 1 : 0;
            }
            sum += (float)cnt * lds_V[k * 33 + u] * lds_upvals[u];
        }
        atomicAdd(&lds_contrib[k], sum);
    }
    __syncthreads();

    if (t < 32) {
        float bias;
        const int dn = lds_dnidx[t];
        if (biasWs) {
            bias = biasWs[dn];
        } else {
            bias = 0.f;
            const float* row = w_enc_down + (size_t)dn * D_SZ;
            for (int d = 0; d < D_SZ; ++d) bias += row[d] * b_dec_up[d];
        }
        out[(size_t)b * 32 + t] = lds_contrib[t] + bias;
    }
}

// ---------------------------------------------------------------------------
extern "C" void kernel_launch(void* const* d_in, const int* in_sizes, int n_in,
                              void* d_out, int out_size, void* d_ws, size_t ws_size,
                              hipStream_t stream) {
    const float* up_vals      = (const float*)d_in[0];
    const float* w_dec_up     = (const float*)d_in[1];
    const float* w_enc_down   = (const float*)d_in[2];
    const float* b_dec_up     = (const float*)d_in[3];
    const int*   up_indices   = (const int*)d_in[4];
    const int*   down_indices = (const int*)d_in[5];
    const int*   connections  = (const int*)d_in[6];
    float*       out          = (float*)d_out;

    const size_t biasBytes = (size_t)F_DOWN * sizeof(float);
    const size_t tBytes    = (size_t)F_UP * D_SZ * sizeof(float);

    float* biasWs = nullptr;
    float* wT     = nullptr;
    char*  ws     = (char*)d_ws;
    if (ws_size >= biasBytes) {
        biasWs = (float*)ws;
        if (ws_size - biasBytes >= tBytes) wT = (float*)(ws + biasBytes);
    }

    if (biasWs)
        scae_bias_kernel<<<F_DOWN / 8, 256, 0, stream>>>(w_enc_down, b_dec_up, biasWs);
    if (wT)
        scae_transpose_kernel<<<dim3(F_UP / 32, D_SZ / 32), 256, 0, stream>>>(w_dec_up, wT);

    scae_main_kernel<<<B_SZ, 256, 0, stream>>>(
        up_vals, w_dec_up, w_enc_down, b_dec_up,
        up_indices, down_indices, connections, out, wT, biasWs);
}